// SpectrumSolver_25280177504693
// MI455X (gfx1250) — compile-verified
//
#include <hip/hip_runtime.h>
#include <math.h>

#define N_X   20000
#define N_ELL 48
#define N_K   1024
#define N_TAU 600

typedef float v2f __attribute__((ext_vector_type(2)));
typedef float v8f __attribute__((ext_vector_type(8)));

#define PI_F 3.14159265358979323846f

// ---------------------------------------------------------------------------
// Stage 1: per-k line-of-sight integrals.
// grid = N_K blocks, 256 threads (8 wave32 = 4 groups of 64 lanes).
// Each block owns one k sample. Per-tau uniform data (trap weight * sources,
// interpolation index/frac) is staged in LDS once; then lane = ell reads
// contiguous 48-float table rows (coalesced, L2-resident) and accumulates.
// Finally U[ell,k] = sqrt(g_k)*Tl, V[ell,k] = sqrt(g_k)*El with
// g_k = (2/pi) * trapw_k * k^2 * P_R(k)  (>= 0), so stage 2 is a pure Gram.
// ---------------------------------------------------------------------------
__global__ __launch_bounds__(256)
void stage1_los(const float* __restrict__ kArr,
                const float* __restrict__ tauArr,
                const float* __restrict__ tau0p,
                const float* __restrict__ S0,
                const float* __restrict__ S1,
                const float* __restrict__ S2,
                const float* __restrict__ SE,
                const float* __restrict__ bx,
                const float* __restrict__ p0,
                const float* __restrict__ p1,
                const float* __restrict__ p2,
                const float* __restrict__ pe,
                const float* __restrict__ Asp,
                const float* __restrict__ nsp,
                float* __restrict__ U,
                float* __restrict__ V)
{
    __shared__ float a0[N_TAU], a1[N_TAU], a2[N_TAU], ae[N_TAU];
    __shared__ int   li0[N_TAU];
    __shared__ float lwf[N_TAU];
    __shared__ float redT[256], redE[256];

    const int bk  = blockIdx.x;
    const int tid = threadIdx.x;

    const float kv     = kArr[bk];
    const float tau0   = tau0p[0];
    const float xmin   = bx[0];
    const float xmax   = bx[N_X - 1];
    const float xscale = (float)(N_X - 1) / (xmax - xmin);

    // Cooperative precompute of all tau-uniform quantities into LDS.
    for (int t = tid; t < N_TAU; t += 256) {
        float tv = tauArr[t];
        float tm = (t > 0)         ? tauArr[t - 1] : tv;
        float tp = (t < N_TAU - 1) ? tauArr[t + 1] : tv;
        float wt = 0.5f * (tp - tm);                 // trapezoid weight in tau
        a0[t] = wt * S0[bk * N_TAU + t];
        a1[t] = wt * S1[bk * N_TAU + t];
        a2[t] = wt * S2[bk * N_TAU + t];
        ae[t] = wt * SE[bk * N_TAU + t];
        float x   = kv * (tau0 - tv);
        float pos = (x - xmin) * xscale;             // uniform-grid interp
        pos = fminf(fmaxf(pos, 0.0f), (float)(N_X - 1));
        int i0 = (int)pos;
        if (i0 > N_X - 2) i0 = N_X - 2;
        li0[t] = i0 * N_ELL;                         // row byte-friendly base
        lwf[t] = pos - (float)i0;
    }
    __syncthreads();

    // lane -> multipole; 4 groups split the tau range for latency hiding.
    const int l6    = tid & 63;
    const int ell   = (l6 < N_ELL) ? l6 : (N_ELL - 1);   // clamp avoids OOB
    const int group = tid >> 6;
    const int tBeg  = group * (N_TAU / 4);
    const int tEnd  = tBeg + (N_TAU / 4);

    float accT = 0.0f, accE = 0.0f;
    for (int t = tBeg; t < tEnd; ++t) {
        const int   base = li0[t];
        const float wf   = lwf[t];
        if (t + 4 < tEnd) {                          // hide the gather latency
            int pfb = li0[t + 4] + ell;
            __builtin_prefetch(&p0[pfb], 0, 1);
            __builtin_prefetch(&p1[pfb], 0, 1);
            __builtin_prefetch(&p2[pfb], 0, 1);
            __builtin_prefetch(&pe[pfb], 0, 1);
        }
        float q0a = p0[base + ell], q0b = p0[base + N_ELL + ell];
        float q1a = p1[base + ell], q1b = p1[base + N_ELL + ell];
        float q2a = p2[base + ell], q2b = p2[base + N_ELL + ell];
        float qea = pe[base + ell], qeb = pe[base + N_ELL + ell];
        float f0 = fmaf(wf, q0b - q0a, q0a);
        float f1 = fmaf(wf, q1b - q1a, q1a);
        float f2 = fmaf(wf, q2b - q2a, q2a);
        float fe = fmaf(wf, qeb - qea, qea);
        accT = fmaf(a0[t], f0, accT);
        accT = fmaf(a1[t], f1, accT);
        accT = fmaf(a2[t], f2, accT);
        accE = fmaf(ae[t], fe, accE);
    }
    redT[tid] = accT;
    redE[tid] = accE;
    __syncthreads();

    if (tid < N_ELL) {
        float sT = redT[tid] + redT[tid + 64] + redT[tid + 128] + redT[tid + 192];
        float sE = redE[tid] + redE[tid + 64] + redE[tid + 128] + redE[tid + 192];
        // k-quadrature weight: g_k = (2/pi) * trapw_k * k^2 * P_R(k) >= 0
        float km = (bk > 0)       ? kArr[bk - 1] : kv;
        float kp = (bk < N_K - 1) ? kArr[bk + 1] : kv;
        float tw = 0.5f * (kp - km);
        float ns = nsp[0], As = Asp[0];
        float PR = As * powf(kv * 20.0f, ns - 1.0f) *
                   (2.0f * PI_F * PI_F) / (kv * kv * kv);
        float g  = (2.0f / PI_F) * tw * kv * kv * PR;
        float sg = sqrtf(fmaxf(g, 0.0f));
        U[tid * N_K + bk] = sg * sT;
        V[tid * N_K + bk] = sg * sE;
    }
}

// ---------------------------------------------------------------------------
// Stage 2: Cl spectra as diagonals of Gram matrices via FP32 WMMA.
//   ClTT = diag(U U^T), ClEE = diag(V V^T), ClTE = diag(U V^T), K = 1024.
// 9 wave32 (one block): wave = 3*spectrum + diag-tile; each wave runs 256
// V_WMMA_F32_16X16X4_F32 ops accumulating its 16x16 tile, then scatters the
// diagonal per the ISA C/D layout (VGPR r holds M = r + 8*(lane>=16)).
// A 16x4 layout: VGPR0 = K{0|2}, VGPR1 = K{1|3} per lane half; B mirrors.
// ---------------------------------------------------------------------------
__global__ __launch_bounds__(288)
void stage2_cl(const float* __restrict__ U,
               const float* __restrict__ V,
               float* __restrict__ out)
{
    const int tid  = threadIdx.x;
    const int wave = tid >> 5;           // 0..8
    const int lane = tid & 31;
    const int s = wave / 3;              // 0=TT, 1=EE, 2=TE
    const int d = wave % 3;              // which diagonal 16x16 tile of 48x48
    const float* __restrict__ X = (s == 1) ? V : U;   // A-operand rows
    const float* __restrict__ Y = (s == 0) ? U : V;   // B-operand rows

    const int m    = lane & 15;
    const int kh   = (lane >> 4) << 1;   // K offset 0 (lanes 0-15) or 2
    const int rowA = (d * 16 + m) * N_K + kh;
    const int rowB = rowA;               // diagonal tile: same row index set

    v8f acc = {};
    for (int kb = 0; kb < N_K; kb += 4) {
        v2f a = *(const v2f*)&X[rowA + kb];   // K = kh, kh+1 (8B aligned)
        v2f b = *(const v2f*)&Y[rowB + kb];
        acc = __builtin_amdgcn_wmma_f32_16x16x4_f32(
                  /*neg_a=*/false, a, /*neg_b=*/false, b,
                  /*c_mod=*/(short)0, acc, /*reuse_a=*/false, /*reuse_b=*/false);
    }

    // Diagonal extraction: VGPR r holds row M = r + 8*(lane>=16), col N = lane%16.
    const int n     = lane & 15;
    const int mbase = (lane < 16) ? 0 : 8;
#pragma unroll
    for (int r = 0; r < 8; ++r) {
        int M = mbase + r;
        if (M == n) out[s * N_ELL + d * 16 + M] = acc[r];
    }
}

// ---------------------------------------------------------------------------
extern "C" void kernel_launch(void* const* d_in, const int* in_sizes, int n_in,
                              void* d_out, int out_size, void* d_ws, size_t ws_size,
                              hipStream_t stream)
{
    const float* k    = (const float*)d_in[0];
    const float* tau  = (const float*)d_in[1];
    const float* tau0 = (const float*)d_in[2];
    const float* S0   = (const float*)d_in[3];
    const float* S1   = (const float*)d_in[4];
    const float* S2   = (const float*)d_in[5];
    const float* SE   = (const float*)d_in[6];
    const float* bx   = (const float*)d_in[7];
    const float* p0   = (const float*)d_in[8];
    const float* p1   = (const float*)d_in[9];
    const float* p2   = (const float*)d_in[10];
    const float* pe   = (const float*)d_in[11];
    const float* As   = (const float*)d_in[12];
    const float* ns   = (const float*)d_in[13];

    float* U = (float*)d_ws;             // [48][1024] sqrt(g_k)*Tl
    float* V = U + N_ELL * N_K;          // [48][1024] sqrt(g_k)*El (384 KB total)

    stage1_los<<<N_K, 256, 0, stream>>>(k, tau, tau0, S0, S1, S2, SE, bx,
                                        p0, p1, p2, pe, As, ns, U, V);
    stage2_cl<<<1, 288, 0, stream>>>(U, V, (float*)d_out);
}